// GNN_6253472383532
// MI455X (gfx1250) — compile-verified
//
#include <hip/hip_runtime.h>
#include <math.h>

#define NBATCH 16384
#define NNODE  12
#define NCH    512
#define STRIDEX (NCH + 4)
#define STATS_BLOCKS 1024
#define STATS_SLAB   (NBATCH / STATS_BLOCKS)
#define BN_EPS 1e-5f

typedef float v2f __attribute__((ext_vector_type(2)));
typedef float v8f __attribute__((ext_vector_type(8)));

// ---------------------------------------------------------------------------
// K0: 512x512 transpose into PAIR-INTERLEAVED layout:
//   out[((c>>1)*512 + d)*2 + (c&1)] = in[d*512 + c]
// so a WMMA B-fragment {W[d][k], W[d][k+1]} is ONE b64 load, coalesced across
// lanes (consecutive d -> consecutive 8B).
// ---------------------------------------------------------------------------
__global__ __launch_bounds__(256) void k_transpose(const float* __restrict__ in,
                                                   float* __restrict__ out) {
  __shared__ float tile[32][33];
  int bx = blockIdx.x & 15;   // input col-tile
  int by = blockIdx.x >> 4;   // input row-tile
  int tx = threadIdx.x & 31;
  int ty = threadIdx.x >> 5;  // 0..7
  int x0 = bx * 32 + tx;
#pragma unroll
  for (int i = 0; i < 32; i += 8) {
    int y = by * 32 + ty + i;
    tile[ty + i][tx] = in[y * 512 + x0];   // tile[a][b] = in[by*32+a][bx*32+b]
  }
  __syncthreads();
  int d = by * 32 + tx;                    // output "row" = original weight row
#pragma unroll
  for (int i = 0; i < 32; i += 8) {
    int c = bx * 32 + ty + i;              // original weight col (k index)
    out[(((size_t)(c >> 1) * 512 + d) << 1) + (c & 1)] = tile[tx][ty + i];
  }
}

// ---------------------------------------------------------------------------
// K1: per-batch dynamic adjacency. One wave per batch.
// sim = X * X^T via V_WMMA_F32_16X16X4_F32 (A-frag == B-frag for a Gram
// matrix). Lanes 12..15 alias rows 0..3 UNGUARDED: the resulting garbage in
// rows/cols 12..15 of sim is finite and never read, and dropping the exec-
// masked loads lets the compiler keep loads in flight across WMMAs.
// ---------------------------------------------------------------------------
__global__ __launch_bounds__(256) void k_adjacency(const float* __restrict__ x,
                                                   float* __restrict__ Aout) {
  __shared__ float simS[8][16][16];
  __shared__ float thrS[8][12];
  __shared__ float dinvS[8][12];
  int wave = threadIdx.x >> 5;
  int lane = threadIdx.x & 31;
  int half = lane >> 4;
  int l16  = lane & 15;
  int bb = blockIdx.x * 8 + wave;
  int rowc = (l16 < NNODE) ? l16 : (l16 - NNODE);   // alias rows 0..3
  const float* xrow = x + (size_t)bb * NNODE * NCH + rowc * NCH + 2 * half;

  v8f acc = {0.f, 0.f, 0.f, 0.f, 0.f, 0.f, 0.f, 0.f};
  v2f a = *(const v2f*)(xrow);              // k = 0
  for (int k = 0; k < NCH; k += 4) {
    int kn = k + 4;
    kn = (kn < NCH) ? kn : 0;               // harmless replay on last iter
    v2f anext = *(const v2f*)(xrow + kn);
    acc = __builtin_amdgcn_wmma_f32_16x16x4_f32(false, a, false, a,
                                                (short)0, acc, false, false);
    a = anext;
  }
#pragma unroll
  for (int v = 0; v < 8; ++v)
    simS[wave][v + 8 * half][l16] = acc[v];
  __syncthreads();

  if (lane < NNODE) {
    float vals[12];
#pragma unroll
    for (int j = 0; j < 12; ++j) vals[j] = simS[wave][lane][j];
    // 4th-largest (tie-correct): max v s.t. #{m : vals[m] >= v} >= 4
    float t4 = -3.4e38f;
#pragma unroll
    for (int j = 0; j < 12; ++j) {
      int cnt = 0;
#pragma unroll
      for (int m = 0; m < 12; ++m) cnt += (vals[m] >= vals[j]) ? 1 : 0;
      if (cnt >= 4 && vals[j] > t4) t4 = vals[j];
    }
    float deg = 0.f;
#pragma unroll
    for (int j = 0; j < 12; ++j) deg += (vals[j] >= t4) ? 1.f : 0.f;
    thrS[wave][lane]  = t4;
    dinvS[wave][lane] = rsqrtf(deg);
  }
  __syncthreads();

  for (int p = lane; p < 144; p += 32) {
    int i = p / 12, j = p % 12;
    float av = (simS[wave][i][j] >= thrS[wave][i]) ? 1.f : 0.f;
    Aout[(size_t)bb * 144 + p] = dinvS[wave][i] * av * dinvS[wave][j];
  }
}

// ---------------------------------------------------------------------------
// K2: fused dual GEMM + graph aggregate. One block (8 waves) per batch.
// X_b staged to LDS (16 rows, 12 real + 4 zero). Each wave owns 64 output
// columns, accumulating BOTH Vx and Ux tiles via fp32 WMMA over K=512.
// All 9 fragment loads of a k-step (1 ds_load_b64 + 8 coalesced
// global_load_b64 from the pair-interleaved weights) are hoisted ahead of the
// 8 WMMAs so loadcnt can drain progressively.
// Then Vx+Vb -> LDS, h = A @ Vx + Ux + Ub -> d_out (h scratch).
// ---------------------------------------------------------------------------
__global__ __launch_bounds__(256) void k_gemm_fused(
    const float* __restrict__ x,
    const float* __restrict__ Uwt, const float* __restrict__ Ub,
    const float* __restrict__ Vwt, const float* __restrict__ Vb,
    const float* __restrict__ Aadj,
    float* __restrict__ h) {
  __shared__ float ldsX[16 * STRIDEX];   // 33.0 KB
  __shared__ float ldsV[12 * STRIDEX];   // 24.8 KB
  __shared__ float ldsA[144];
  int b = blockIdx.x;
  int tid = threadIdx.x;
  const float* xb = x + (size_t)b * NNODE * NCH;
  for (int idx = tid; idx < 16 * NCH; idx += 256) {
    int r  = idx >> 9;
    int cc = idx & 511;
    ldsX[r * STRIDEX + cc] = (r < NNODE) ? xb[r * NCH + cc] : 0.f;
  }
  for (int idx = tid; idx < 144; idx += 256)
    ldsA[idx] = Aadj[(size_t)b * 144 + idx];
  __syncthreads();

  int wave = tid >> 5;
  int lane = tid & 31;
  int half = lane >> 4;
  int l16  = lane & 15;
  int colbase = wave * 64;

  const v2f* Vw2 = (const v2f*)Vwt;   // [256][512] pair-interleaved
  const v2f* Uw2 = (const v2f*)Uwt;

  v8f z8 = {0.f, 0.f, 0.f, 0.f, 0.f, 0.f, 0.f, 0.f};
  v8f accV[4] = {z8, z8, z8, z8};
  v8f accU[4] = {z8, z8, z8, z8};

  for (int k = 0; k < NCH; k += 4) {
    int kk = k + 2 * half;              // even
    int kh = kk >> 1;                   // pair row in Wt2
    v2f a = *(const v2f*)(&ldsX[l16 * STRIDEX + kk]);
    v2f bv[4], bu[4];
#pragma unroll
    for (int t = 0; t < 4; ++t) {
      int d = colbase + 16 * t + l16;
      bv[t] = Vw2[(size_t)kh * 512 + d];
      bu[t] = Uw2[(size_t)kh * 512 + d];
    }
#pragma unroll
    for (int t = 0; t < 4; ++t) {
      accV[t] = __builtin_amdgcn_wmma_f32_16x16x4_f32(false, a, false, bv[t],
                                                      (short)0, accV[t], false, false);
      accU[t] = __builtin_amdgcn_wmma_f32_16x16x4_f32(false, a, false, bu[t],
                                                      (short)0, accU[t], false, false);
    }
  }

  // Vx + Vb into LDS (only the 12 real rows)
#pragma unroll
  for (int t = 0; t < 4; ++t) {
    int col = colbase + 16 * t + l16;
    float vb = Vb[col];
#pragma unroll
    for (int v = 0; v < 8; ++v) {
      int row = v + 8 * half;
      if (row < NNODE)
        ldsV[row * STRIDEX + col] = accV[t][v] + vb;
    }
  }
  __syncthreads();

  // h = A @ Vx + Ux + Ub
  float* hb = h + (size_t)b * NNODE * NCH;
#pragma unroll
  for (int t = 0; t < 4; ++t) {
    int col = colbase + 16 * t + l16;
    float ub = Ub[col];
#pragma unroll
    for (int v = 0; v < 8; ++v) {
      int row = v + 8 * half;
      if (row < NNODE) {
        float s = accU[t][v] + ub;
#pragma unroll
        for (int j = 0; j < NNODE; ++j)
          s += ldsA[row * 12 + j] * ldsV[j * STRIDEX + col];
        hb[row * NCH + col] = s;
      }
    }
  }
}

// ---------------------------------------------------------------------------
// K3: per-node partial sums / sums-of-squares, deterministic tree reduction.
// ---------------------------------------------------------------------------
__global__ __launch_bounds__(256) void k_stats_partial(const float* __restrict__ h,
                                                       float* __restrict__ partial) {
  __shared__ float r1[256];
  __shared__ float r2[256];
  int tid = threadIdx.x;
  int b0 = blockIdx.x * STATS_SLAB;
#pragma unroll 1
  for (int n = 0; n < NNODE; ++n) {
    float s = 0.f, s2 = 0.f;
    for (int bb = 0; bb < STATS_SLAB; ++bb) {
      const float* row = h + ((size_t)(b0 + bb) * NNODE + n) * NCH;
      for (int cc = tid; cc < NCH; cc += 256) {
        float val = row[cc];
        s += val; s2 += val * val;
      }
    }
    r1[tid] = s; r2[tid] = s2;
    __syncthreads();
    for (int off = 128; off > 0; off >>= 1) {
      if (tid < off) { r1[tid] += r1[tid + off]; r2[tid] += r2[tid + off]; }
      __syncthreads();
    }
    if (tid == 0) {
      partial[(size_t)blockIdx.x * 24 + n * 2]     = r1[0];
      partial[(size_t)blockIdx.x * 24 + n * 2 + 1] = r2[0];
    }
    __syncthreads();
  }
}

// ---------------------------------------------------------------------------
// K4: finalize BN stats, fold gamma/beta into (scale, shift). Deterministic.
// ---------------------------------------------------------------------------
__global__ __launch_bounds__(64) void k_stats_final(const float* __restrict__ partial,
                                                    const float* __restrict__ gamma,
                                                    const float* __restrict__ beta,
                                                    float* __restrict__ stats) {
  int n = threadIdx.x;
  if (n < NNODE) {
    float s = 0.f, s2 = 0.f;
    for (int p = 0; p < STATS_BLOCKS; ++p) {
      s  += partial[(size_t)p * 24 + n * 2];
      s2 += partial[(size_t)p * 24 + n * 2 + 1];
    }
    const float cnt = (float)((size_t)NBATCH * NCH);
    float mean = s / cnt;
    float var  = s2 / cnt - mean * mean;
    float rstd = rsqrtf(var + BN_EPS);
    float scale = gamma[n] * rstd;
    float shift = beta[n] - mean * scale;
    stats[n]         = scale;
    stats[NNODE + n] = shift;
  }
}

// ---------------------------------------------------------------------------
// K5: out = relu(x + h*scale[n] + shift[n]), float4-vectorized, in place on h.
// ---------------------------------------------------------------------------
__global__ __launch_bounds__(256) void k_finalize(const float* __restrict__ x,
                                                  const float* __restrict__ stats,
                                                  float* __restrict__ h) {
  __shared__ float sc[NNODE], sh[NNODE];
  if (threadIdx.x < NNODE) {
    sc[threadIdx.x] = stats[threadIdx.x];
    sh[threadIdx.x] = stats[NNODE + threadIdx.x];
  }
  __syncthreads();
  const float4* x4 = (const float4*)x;
  float4* h4 = (float4*)h;
  size_t total4 = (size_t)NBATCH * NNODE * NCH / 4;
  size_t stride = (size_t)gridDim.x * blockDim.x;
  for (size_t i = (size_t)blockIdx.x * 256 + threadIdx.x; i < total4; i += stride) {
    int n = (int)((i >> 7) % NNODE);   // 128 float4 per (b, n) row
    float4 xv = x4[i];
    float4 hv = h4[i];
    float scale = sc[n], shift = sh[n];
    float4 o;
    o.x = fmaxf(0.f, xv.x + hv.x * scale + shift);
    o.y = fmaxf(0.f, xv.y + hv.y * scale + shift);
    o.z = fmaxf(0.f, xv.z + hv.z * scale + shift);
    o.w = fmaxf(0.f, xv.w + hv.w * scale + shift);
    h4[i] = o;
  }
}

// ---------------------------------------------------------------------------
extern "C" void kernel_launch(void* const* d_in, const int* in_sizes, int n_in,
                              void* d_out, int out_size, void* d_ws, size_t ws_size,
                              hipStream_t stream) {
  (void)in_sizes; (void)n_in; (void)out_size; (void)ws_size;
  const float* x     = (const float*)d_in[0];
  const float* Uw    = (const float*)d_in[1];
  const float* Ub    = (const float*)d_in[2];
  const float* Vw    = (const float*)d_in[3];
  const float* Vb    = (const float*)d_in[4];
  const float* gamma = (const float*)d_in[5];
  const float* beta  = (const float*)d_in[6];
  float* h = (float*)d_out;   // d_out doubles as scratch for h, finalized in place

  float* ws      = (float*)d_ws;
  float* Aadj    = ws;                               // 16384*144 floats (9.4 MB)
  float* Uwt     = Aadj + (size_t)NBATCH * 144;      // 512*512 (pair-interleaved)
  float* Vwt     = Uwt + (size_t)512 * 512;          // 512*512 (pair-interleaved)
  float* partial = Vwt + (size_t)512 * 512;          // 1024*24
  float* stats   = partial + (size_t)STATS_BLOCKS * 24;  // 24

  k_transpose<<<256, 256, 0, stream>>>(Uw, Uwt);
  k_transpose<<<256, 256, 0, stream>>>(Vw, Vwt);
  k_adjacency<<<NBATCH / 8, 256, 0, stream>>>(x, Aadj);
  k_gemm_fused<<<NBATCH, 256, 0, stream>>>(x, Uwt, Ub, Vwt, Vb, Aadj, h);
  k_stats_partial<<<STATS_BLOCKS, 256, 0, stream>>>(h, partial);
  k_stats_final<<<1, 64, 0, stream>>>(partial, gamma, beta, stats);
  k_finalize<<<4096, 256, 0, stream>>>(x, stats, h);
}